// SpikeFP32SiLU_29875792511512
// MI455X (gfx1250) — compile-verified
//
#include <hip/hip_runtime.h>
#include <stdint.h>

// SpikeFP32SiLU for MI455X (gfx1250): pure HBM-streaming elementwise op.
// 1 GiB in + 1 GiB out @ 23.3 TB/s => ~92us roofline floor. Strategy:
// async-to-LDS (coalesced b128, lane-contiguous), double-buffered, padded
// LDS layout for conflict-free ds_load_b128, nontemporal b128 stores.

typedef float v4f __attribute__((ext_vector_type(4)));

#define TPB            256
#define VALS_TILE      256                      // one value per thread per tile
#define VAL_BYTES      128                      // 32 spike floats per value
#define TILE_BYTES     (VALS_TILE * VAL_BYTES)  // 32768 B global tile
#define LDS_VAL_STRIDE 144                      // padded stride (bytes): 36 words -> bank-friendly
#define LDS_TILE_BYTES (VALS_TILE * LDS_VAL_STRIDE)  // 36864 B
#define LDS_TILE_FLTS  (LDS_TILE_BYTES / 4)          // 9216 floats

__device__ __forceinline__ unsigned lds_byte_off(const void* p) {
  // low 32 bits of a generic pointer into LDS == wave-relative LDS byte address
  return (unsigned)(size_t)p;
}

// Issue 8x global_load_async_to_lds_b128 covering one 32KB tile.
// Instruction k: lane t copies 16B from global [tile_base + k*4096 + t*16]
// (lane-contiguous => fully coalesced 512B/instruction) into LDS at a
// transposed location so value v's 32 floats are contiguous at stride 144B:
//   global chunk (k*4096 + t*16) belongs to value (k*32 + t/8), sub-chunk (t%8).
__device__ __forceinline__ void issue_tile_async(const float* __restrict__ xbase,
                                                 unsigned tile_byte_base,
                                                 unsigned lds_buf_base,
                                                 int t) {
  unsigned voff = tile_byte_base + (unsigned)t * 16u;
  unsigned lds0 = lds_buf_base + (unsigned)(t >> 3) * (unsigned)LDS_VAL_STRIDE
                               + (unsigned)(t & 7) * 16u;
#pragma unroll
  for (int k = 0; k < 8; ++k) {
    asm volatile("global_load_async_to_lds_b128 %0, %1, %2"
                 :
                 : "v"(lds0 + (unsigned)k * (32u * (unsigned)LDS_VAL_STRIDE)),
                   "v"(voff + (unsigned)k * 4096u),
                   "s"(xbase)
                 : "memory");
  }
}

__device__ __forceinline__ uint32_t pack32(const v4f* p) {
  // spike floats are exactly 0.0f or 1.0f; bit = (bits(f) >> 23) & 1.
  // MSB-first memory order -> Horner accumulate.
  uint32_t u = 0;
#pragma unroll
  for (int k = 0; k < 8; ++k) {
    v4f f = p[k];
    u = (u << 4)
      | (((__float_as_uint(f.x) >> 23) & 1u) << 3)
      | (((__float_as_uint(f.y) >> 23) & 1u) << 2)
      | (((__float_as_uint(f.z) >> 23) & 1u) << 1)
      |  ((__float_as_uint(f.w) >> 23) & 1u);
  }
  return u;
}

__device__ __forceinline__ void unpack_store(uint32_t yu, float* __restrict__ op) {
#pragma unroll
  for (int k = 0; k < 8; ++k) {
    v4f o;
    o.x = (float)((yu >> (31 - 4 * k)) & 1u);
    o.y = (float)((yu >> (30 - 4 * k)) & 1u);
    o.z = (float)((yu >> (29 - 4 * k)) & 1u);
    o.w = (float)((yu >> (28 - 4 * k)) & 1u);
    __builtin_nontemporal_store(o, (v4f*)(op + 4 * k));  // streamed-once: don't pollute L2
  }
}

__device__ __forceinline__ float silu(float x) {
  return x / (1.0f + __expf(-x));
}

__global__ __launch_bounds__(TPB)
void spike_silu_kernel(const float* __restrict__ x, float* __restrict__ out,
                       long nvals, int ntiles) {
  __shared__ __align__(16) float smem[2 * LDS_TILE_FLTS];
  const int t = threadIdx.x;
  const int G = gridDim.x;
  const unsigned lbase0 = lds_byte_off(&smem[0]);
  const unsigned lbase1 = lds_byte_off(&smem[LDS_TILE_FLTS]);

  int tile = blockIdx.x;
  if (tile < ntiles) {
    issue_tile_async(x, (unsigned)tile * (unsigned)TILE_BYTES, lbase0, t);
    int cur = 0;
    for (;;) {
      const int nxt = tile + G;
      const bool more = (nxt < ntiles);
      if (more) {
        // prefetch next tile into the other buffer (safe: trailing barrier of
        // the previous iteration guarantees everyone finished reading it)
        issue_tile_async(x, (unsigned)nxt * (unsigned)TILE_BYTES,
                         cur ? lbase0 : lbase1, t);
        asm volatile("s_wait_asynccnt 0x8" ::: "memory");  // tile `tile` done (in-order)
      } else {
        asm volatile("s_wait_asynccnt 0x0" ::: "memory");
      }
      __syncthreads();  // make all waves' async LDS writes visible

      const float* lbuf = &smem[cur ? LDS_TILE_FLTS : 0];
      const v4f* p = (const v4f*)(lbuf + t * (LDS_VAL_STRIDE / 4));
      uint32_t u = pack32(p);
      uint32_t yu = __float_as_uint(silu(__uint_as_float(u)));
      const long vidx = (long)tile * VALS_TILE + t;
      unpack_store(yu, out + vidx * 32);

      __syncthreads();  // done reading this buffer before it is overwritten
      if (!more) break;
      tile = nxt;
      cur ^= 1;
    }
  }

  // Tail (nvals % 256 != 0): direct global path. (Exact multiple for the
  // reference shapes, so this normally never executes.)
  const long rem = (long)ntiles * VALS_TILE;
  for (long v = rem + (long)blockIdx.x * TPB + t; v < nvals; v += (long)G * TPB) {
    const v4f* ip = (const v4f*)(x + v * 32);
    uint32_t u = pack32(ip);
    uint32_t yu = __float_as_uint(silu(__uint_as_float(u)));
    unpack_store(yu, out + v * 32);
  }
}

extern "C" void kernel_launch(void* const* d_in, const int* in_sizes, int n_in,
                              void* d_out, int out_size, void* d_ws, size_t ws_size,
                              hipStream_t stream) {
  (void)n_in; (void)out_size; (void)d_ws; (void)ws_size;
  const float* x = (const float*)d_in[0];
  float* out = (float*)d_out;

  const long nbits = (long)in_sizes[0];        // 268,435,456 for reference shapes
  const long nvals = nbits / 32;               // 8,388,608 values
  const int ntiles = (int)(nvals / VALS_TILE); // 32,768 full tiles

  int grid = ntiles;
  if (grid < 1) grid = (int)((nvals + TPB - 1) / TPB);
  if (grid < 1) grid = 1;
  if (grid > 4096) grid = 4096;                // grid-stride: >=8 tiles/block keeps the
                                               // async double-buffer pipeline busy
  spike_silu_kernel<<<grid, TPB, 0, stream>>>(x, out, nvals, ntiles);
}